// EdgeToAtomLayer_21191368639075
// MI455X (gfx1250) — compile-verified
//
#include <hip/hip_runtime.h>

#define WAVES_PER_BLOCK 8
#define BLOCK_THREADS (WAVES_PER_BLOCK * 32)
#define FEAT_D 64                      // floats per edge row
#define ROW_BYTES (FEAT_D * 4)         // 256 B per edge row

// ---------------------------------------------------------------------------
// Scatter-add: vi[idx[e], :] += mj[e, :]
//   - 16 lanes per edge, each lane moves one float4 (16B) of the 256B row
//   - per-wave double-buffered async global->LDS staging (gfx1250 path:
//     global_load_async_to_lds_b128 th:TH_LOAD_NT + s_wait_asynccnt); the NT
//     hint keeps the 256MB read-once stream from evicting the L2-resident
//     accumulators (25.6MB out + 4MB indices stay hot in the 192MB L2)
//   - unsafeAtomicAdd -> no-return global_atomic_add_f32, resolves in L2
//   - all offsets 32-bit (256MB / 25.6MB extents) -> minimal address VALU
// ---------------------------------------------------------------------------
__device__ __forceinline__ void async_load_row16(unsigned lds_byte_off,
                                                 const void* gaddr) {
    // VDST = LDS byte offset (per-lane VGPR), VADDR = 64-bit global address
    asm volatile("global_load_async_to_lds_b128 %0, %1, off th:TH_LOAD_NT"
                 :
                 : "v"(lds_byte_off), "v"(gaddr)
                 : "memory");
}

__device__ __forceinline__ void wait_async_le(int n) {
    if (n == 0)
        asm volatile("s_wait_asynccnt 0" ::: "memory");
    else
        asm volatile("s_wait_asynccnt 1" ::: "memory");
}

__global__ __launch_bounds__(BLOCK_THREADS) void
edge_scatter_add_kernel(const float* __restrict__ mj,
                        const int* __restrict__ tgt,  // target row: ei + E
                        float* __restrict__ out,
                        int npairs) {                 // E/2 (E even)
    // Per-wave staging: [wave][buf(2)][edge(2)][col(16)] float4  -> 8KB total
    __shared__ float4 stage[WAVES_PER_BLOCK * 2 * 2 * 16];

    const int lane = threadIdx.x & 31;
    const int wave = threadIdx.x >> 5;
    const int sub  = lane >> 4;   // which edge of the wave's pair (0/1)
    const int col  = lane & 15;   // which float4 of the 64-float row

    const int total_waves = (int)gridDim.x * WAVES_PER_BLOCK;
    int p = (int)blockIdx.x * WAVES_PER_BLOCK + wave;   // edge-pair index

    const char* __restrict__ mjb = (const char*)mj;
    char* __restrict__ outb = (char*)out;

    float4* mybuf = &stage[wave * 64];
    // addrspace(3) offset: generic LDS pointers carry the LDS byte offset in
    // their low 32 bits (ISA 10.2 aperture mapping), so truncation is exact.
    const unsigned lds_wave_base = (unsigned)(size_t)mybuf;
    const unsigned lane_slot = (unsigned)(sub * 16 + col) * 16u;  // bytes in buf

    auto issue = [&](int pp, int buf) {
        const unsigned e = 2u * (unsigned)pp + (unsigned)sub;
        const unsigned goff = e * ROW_BYTES + (unsigned)col * 16u;
        async_load_row16(lds_wave_base + (unsigned)buf * 512u + lane_slot,
                         mjb + goff);
    };

    int buf = 0;
    if (p < npairs) {
        issue(p, buf);
        while (p < npairs) {
            const int pn = p + total_waves;
            const bool more = (pn < npairs);            // wave-uniform
            if (more) {
                issue(pn, buf ^ 1);
                wait_async_le(1);                       // stage for p is done
            } else {
                wait_async_le(0);
            }

            // Consume current stage
            const int e = 2 * p + sub;
            const int idx = tgt[e];                     // target node
            const float4 v = mybuf[buf * 32 + sub * 16 + col];
            float* dst = (float*)(outb + (unsigned)idx * ROW_BYTES +
                                  (unsigned)col * 16u);
            unsafeAtomicAdd(dst + 0, v.x);   // global_atomic_add_f32 (no-return)
            unsafeAtomicAdd(dst + 1, v.y);
            unsafeAtomicAdd(dst + 2, v.z);
            unsafeAtomicAdd(dst + 3, v.w);

            p = pn;
            buf ^= 1;
        }
    }
}

// ---------------------------------------------------------------------------
// Zero-fill the output accumulators (harness poisons d_out with 0xAA).
// ---------------------------------------------------------------------------
__global__ void zero_f4_kernel(float4* __restrict__ out, int n4) {
    int i = blockIdx.x * blockDim.x + threadIdx.x;
    int stride = gridDim.x * blockDim.x;
    float4 z = make_float4(0.f, 0.f, 0.f, 0.f);
    for (; i < n4; i += stride) out[i] = z;
}

// ---------------------------------------------------------------------------
// Launch
// ---------------------------------------------------------------------------
extern "C" void kernel_launch(void* const* d_in, const int* in_sizes, int n_in,
                              void* d_out, int out_size, void* d_ws, size_t ws_size,
                              hipStream_t stream) {
    const float* mj = (const float*)d_in[0];
    const int*   ei = (const int*)d_in[1];
    float* out = (float*)d_out;

    const int E  = in_sizes[0] / FEAT_D;   // 1,000,000 (even)
    const int n4 = out_size / 4;           // N*D/4 float4s

    // 1) zero the accumulators
    int zb = (n4 + BLOCK_THREADS - 1) / BLOCK_THREADS;
    if (zb > 4096) zb = 4096;
    if (zb < 1) zb = 1;
    zero_f4_kernel<<<zb, BLOCK_THREADS, 0, stream>>>((float4*)out, n4);

    // 2) scatter-add with async-staged streaming reads
    const int blocks = 2048;  // 16384 waves; ~30 edge-pairs per wave
    edge_scatter_add_kernel<<<blocks, BLOCK_THREADS, 0, stream>>>(
        mj, ei + E, out, E / 2);
}